// EncoderTransformer_35631048688190
// MI455X (gfx1250) — compile-verified
//
#include <hip/hip_runtime.h>
#include <math.h>

#define B_ 64
#define C_ 4
#define N_ 32
#define W_ 32
#define D_ 512
#define DK_ 64
#define VOCAB_ 32000

typedef float v2f __attribute__((ext_vector_type(2)));
typedef float v8f __attribute__((ext_vector_type(8)));

// ---------------------------------------------------------------------------
// Kernel 0: swizzle W[512,64] (row-major) into WMMA f32 16x16x4 B-fragment
// order.  Fragment f = (nt*128 + kk4)*32 + lane holds two floats:
//   j=0,1 : W[(kk4*4 + (lane<16 ? 0 : 2) + j)*64 + nt*16 + (lane&15)]
// so each wave K-step becomes one coalesced 256B global_load_b64.
// ---------------------------------------------------------------------------
__global__ void swizzleB(const float* __restrict__ Wsrc, float* __restrict__ Wsw) {
    int tid = blockIdx.x * blockDim.x + threadIdx.x;        // 0..16383
    if (tid >= 4 * 128 * 32) return;
    int lane = tid & 31;
    int kk4  = (tid >> 5) & 127;
    int nt   = tid >> 12;
    int kbase = kk4 * 4 + ((lane & 16) ? 2 : 0);
    int col   = nt * 16 + (lane & 15);
    float2 f;
    f.x = Wsrc[(kbase + 0) * DK_ + col];
    f.y = Wsrc[(kbase + 1) * DK_ + col];
    *(float2*)&Wsw[(size_t)tid * 2] = f;
}

// ---------------------------------------------------------------------------
// Kernel 1: Out[M,64] = A[M,512] @ W (+bias) via V_WMMA_F32_16X16X4_F32.
// 256 threads = 8 waves; wave (mt,nt) owns 16x16 tile of a 32x64 block.
// A block staged in LDS with row stride 516 floats -> bank-conflict-free
// b64 fragment reads (bank = (4*row + k) mod 64 distinct across the wave).
// ---------------------------------------------------------------------------
#define LDSA_STRIDE 516
__global__ __launch_bounds__(256) void gemm_wmma(
    const float* __restrict__ A, const float* __restrict__ Bsw,
    const float* __restrict__ bias, float* __restrict__ Out)
{
    __shared__ float As[32 * LDSA_STRIDE];
    const int tid = threadIdx.x;
    const int rowbase = blockIdx.x * 32;

    // stage 32x512 f32 block of A (coalesced float4 loads)
    #pragma unroll
    for (int i = 0; i < 16; ++i) {
        int idx = i * 256 + tid;
        int r  = idx >> 7;           // /128
        int c4 = (idx & 127) << 2;   // *4
        float4 val = *(const float4*)&A[(size_t)(rowbase + r) * D_ + c4];
        *(float4*)&As[r * LDSA_STRIDE + c4] = val;
    }
    __syncthreads();

    const int wave = tid >> 5;           // 0..7
    const int lane = tid & 31;
    const int mt = wave >> 2;            // 0..1
    const int nt = wave & 3;             // 0..3
    const int mrow = mt * 16 + (lane & 15);
    const int ksel = (lane & 16) ? 2 : 0;

    v8f acc = {};
    const float* aptr = &As[mrow * LDSA_STRIDE + ksel];
    const float* bptr = Bsw + ((size_t)nt * 128 * 32 + lane) * 2;

    #pragma unroll 4
    for (int kk4 = 0; kk4 < 128; ++kk4) {
        v2f a = *(const v2f*)(aptr + kk4 * 4);   // LDS b64, conflict-free
        v2f b = *(const v2f*)(bptr + kk4 * 64);  // global b64, coalesced
        acc = __builtin_amdgcn_wmma_f32_16x16x4_f32(
                  false, a, false, b, (short)0, acc, false, false);
    }

    // D layout: lane -> col (lane&15); VGPR j -> row j (+8 for lanes 16..31)
    const int ncol  = nt * 16 + (lane & 15);
    const float bv  = bias ? bias[ncol] : 0.0f;
    const int rbase = rowbase + mt * 16 + ((lane & 16) ? 8 : 0);
    #pragma unroll
    for (int j = 0; j < 8; ++j)
        Out[(size_t)(rbase + j) * DK_ + ncol] = acc[j] + bv;
}

// ---------------------------------------------------------------------------
// Kernel 2: per-bag attention.  emb_pre already contains emb@W_pre + b_pre,
// and emb_pre[0] == b_pre (since emb[PAD]=0) == exactly what the reference's
// zeroed bag rows yield, so words >= len gather row 0.  Softmax over all 32
// positions (unmasked), matching the reference.
// Context phase uses b128 gathers: threads 0..127 accumulate words 0..15,
// threads 128..255 accumulate words 16..31 over the same float4 d-slice;
// halves are merged through LDS and stored with b128.
// ---------------------------------------------------------------------------
__global__ __launch_bounds__(256) void attn_kernel(
    const int*   __restrict__ token_ids,  const int* __restrict__ node_lengths,
    const int*   __restrict__ node_sizes, const int* __restrict__ cross_lengths,
    const float* __restrict__ emb,        const float* __restrict__ emb_pre,
    const float* __restrict__ q_all,      const float* __restrict__ v,
    float* __restrict__ out)
{
    const int t   = blockIdx.x;
    const int tid = threadIdx.x;
    const int b = t / (C_ * N_);
    const int c = (t / N_) % C_;
    const int n = t % N_;

    float* outp = out + (size_t)t * D_;
    const bool masked = (n >= node_sizes[b * C_ + c]) || (c >= cross_lengths[b]);
    if (masked) {                         // block-uniform branch
        float2 z = {0.f, 0.f};
        *(float2*)&outp[tid * 2] = z;
        return;
    }

    __shared__ int    sTok[W_];
    __shared__ float  sE[W_];
    __shared__ float  sScore[W_];
    __shared__ float4 sAcc[128];

    const int len = node_lengths[t];
    if (tid < W_) {
        int tok = token_ids[t * W_ + tid];
        sTok[tid] = (tid < len) ? tok : 0;    // PAD row beyond length
    }
    __syncthreads();

    // energy: 8 threads per word, 8 k-values each
    {
        const int w  = tid >> 3;
        const int kk = (tid & 7) * 8;
        const float* pp = emb_pre + (size_t)sTok[w] * DK_ + kk;
        const float* qq = q_all + b * DK_ + kk;
        float partial = 0.f;
        #pragma unroll
        for (int j = 0; j < 8; ++j)
            partial += tanhf(pp[j] + qq[j]) * v[kk + j];
        partial += __shfl_xor(partial, 1);
        partial += __shfl_xor(partial, 2);
        partial += __shfl_xor(partial, 4);
        if ((tid & 7) == 0) sE[w] = partial;
    }
    __syncthreads();

    // softmax over all 32 words (wave 0)
    if (tid < W_) {
        float e = sE[tid];
        float m = e;
        for (int off = 1; off < 32; off <<= 1)
            m = fmaxf(m, __shfl_xor(m, off));
        float ex = expf(e - m);
        float s = ex;
        for (int off = 1; off < 32; off <<= 1)
            s += __shfl_xor(s, off);
        sScore[tid] = ex / s;
    }
    __syncthreads();

    // context: score-weighted gather of emb rows (L2-resident), b128 loads
    const int dq = (tid & 127) * 4;       // float4 d-slice
    const int h  = tid >> 7;              // word half: 0 -> w 0..15, 1 -> w 16..31
    float4 acc = {0.f, 0.f, 0.f, 0.f};
    const int wbeg = h * 16;
    const int wend = (len < wbeg + 16) ? len : (wbeg + 16);
    for (int w = wbeg; w < wend; ++w) {
        const float  s = sScore[w];
        const float4 e = *(const float4*)(emb + (size_t)sTok[w] * D_ + dq);
        acc.x = fmaf(s, e.x, acc.x);
        acc.y = fmaf(s, e.y, acc.y);
        acc.z = fmaf(s, e.z, acc.z);
        acc.w = fmaf(s, e.w, acc.w);
    }
    if (h == 1) sAcc[tid & 127] = acc;
    __syncthreads();
    if (h == 0) {
        const float4 o = sAcc[tid];
        acc.x += o.x; acc.y += o.y; acc.z += o.z; acc.w += o.w;
        *(float4*)&outp[dq] = acc;        // b128 store
    }
}

// ---------------------------------------------------------------------------
extern "C" void kernel_launch(void* const* d_in, const int* in_sizes, int n_in,
                              void* d_out, int out_size, void* d_ws, size_t ws_size,
                              hipStream_t stream) {
    const int*   token_ids     = (const int*)d_in[0];
    const int*   node_lengths  = (const int*)d_in[1];
    const int*   node_sizes    = (const int*)d_in[2];
    const int*   cross_lengths = (const int*)d_in[3];
    const float* con_hidden    = (const float*)d_in[4];
    const float* emb           = (const float*)d_in[5];
    const float* W_pre         = (const float*)d_in[6];
    const float* b_pre         = (const float*)d_in[7];
    const float* W_q           = (const float*)d_in[8];
    const float* vvec          = (const float*)d_in[9];

    float* ws      = (float*)d_ws;
    float* Wpre_sw = ws;                        // 32768 floats (128 KB)
    float* Wq_sw   = ws + 32768;                // 32768 floats
    float* emb_pre = ws + 65536;                // 32000*64 floats (8 MB)
    float* q_all   = emb_pre + (size_t)VOCAB_ * DK_;  // 64*64 floats

    swizzleB<<<64, 256, 0, stream>>>(W_pre, Wpre_sw);
    swizzleB<<<64, 256, 0, stream>>>(W_q,   Wq_sw);

    gemm_wmma<<<VOCAB_ / 32, 256, 0, stream>>>(emb,        Wpre_sw, b_pre,   emb_pre);
    gemm_wmma<<<B_ / 32,     256, 0, stream>>>(con_hidden, Wq_sw,   nullptr, q_all);

    attn_kernel<<<B_ * C_ * N_, 256, 0, stream>>>(
        token_ids, node_lengths, node_sizes, cross_lengths,
        emb, emb_pre, q_all, vvec, (float*)d_out);

    // second output: hidden = con_hidden (identity reshape)
    hipMemcpyAsync((float*)d_out + (size_t)B_ * C_ * N_ * D_, con_hidden,
                   (size_t)B_ * D_ * sizeof(float),
                   hipMemcpyDeviceToDevice, stream);
}